// BATPredictor_90254442758600
// MI455X (gfx1250) — compile-verified
//
#include <hip/hip_runtime.h>
#include <hip/hip_bf16.h>
#include <math.h>

typedef __attribute__((ext_vector_type(2))) float v2f;
typedef __attribute__((ext_vector_type(4))) float v4f;
typedef __attribute__((ext_vector_type(8))) float v8f;

#define B_DIM 16
#define T_DIM 1500
#define C_DIM 512
#define TILES_PER_B 94          // ceil(1500/16)
#define LDSROW 516              // 512 + 4 pad: 16B-aligned rows, bank-conflict-free

// kernel-4 streaming tile: 12 rows x 512 ch x 4B = 24KB, 125 tiles = 1500 rows
#define TT 12
#define NTILES 125

// ---------------------------------------------------------------------------
// Kernel 1: fused depthwise-conv3 + residual + ReLU + GEMV(lin_w) + sigmoid
// One wave (32 threads) per 16-t tile; K marched over C via v_wmma_f32_16x16x4_f32.
// ---------------------------------------------------------------------------
__global__ __launch_bounds__(32) void alphas_kernel(
    const float* __restrict__ hidden, const float* __restrict__ conv_w,
    const float* __restrict__ conv_b, const float* __restrict__ lin_w,
    const float* __restrict__ lin_b, float* __restrict__ alphas)
{
  __shared__ float tile[18 * LDSROW];
  __shared__ float s_w0[C_DIM], s_w1[C_DIM], s_w2[C_DIM], s_bc[C_DIM], s_lw[C_DIM];

  const int wg   = blockIdx.x;
  const int b    = wg / TILES_PER_B;
  const int t0   = (wg % TILES_PER_B) * 16;
  const int lane = threadIdx.x;
  const float* hb = hidden + (size_t)b * T_DIM * C_DIM;

  // stage weights (coalesced)
  for (int i = lane; i < C_DIM; i += 32) {
    s_w0[i] = conv_w[i * 3 + 0];
    s_w1[i] = conv_w[i * 3 + 1];
    s_w2[i] = conv_w[i * 3 + 2];
    s_bc[i] = conv_b[i];
    s_lw[i] = lin_w[i];
  }
  // stage rows t0-1 .. t0+16 of hidden (zero-pad outside [0,T))
  for (int r = 0; r < 18; ++r) {
    const int tg = t0 - 1 + r;
    if (tg >= 0 && tg < T_DIM) {
      const float* src = hb + (size_t)tg * C_DIM;
      for (int i = lane * 4; i < C_DIM; i += 128) {
        *(v4f*)&tile[r * LDSROW + i] = *(const v4f*)(src + i);
      }
    } else {
      for (int i = lane; i < C_DIM; i += 32) tile[r * LDSROW + i] = 0.0f;
    }
  }
  __syncthreads();

  // lane -> (m, k-half): m = lane%16 (row t0+m), k = 2*(lane/16) + {0,1}
  const int tl    = lane & 15;
  const int khalf = lane >> 4;
  const float* rm1 = &tile[(tl + 0) * LDSROW];  // t-1
  const float* r0  = &tile[(tl + 1) * LDSROW];  // t
  const float* rp1 = &tile[(tl + 2) * LDSROW];  // t+1

  v8f acc = {};
  #pragma unroll 4
  for (int k0 = 0; k0 < C_DIM; k0 += 4) {
    const int c = k0 + 2 * khalf;
    float a0 = s_w0[c]     * rm1[c]     + s_w1[c]     * r0[c]     + s_w2[c]     * rp1[c]
             + s_bc[c]     + r0[c];
    float a1 = s_w0[c + 1] * rm1[c + 1] + s_w1[c + 1] * r0[c + 1] + s_w2[c + 1] * rp1[c + 1]
             + s_bc[c + 1] + r0[c + 1];
    a0 = fmaxf(a0, 0.0f);
    a1 = fmaxf(a1, 0.0f);
    v2f A = {a0, a1};
    v2f Bv = {s_lw[c], s_lw[c + 1]};         // depends only on k -> every N column equal
    acc = __builtin_amdgcn_wmma_f32_16x16x4_f32(
        false, A, false, Bv, (short)0, acc, false, false);
  }

  // D layout: lane n<16 holds M=0..7 in acc[0..7]; lane n>=16 holds M=8..15.
  const float lb = lin_b[0];
  if (lane == 0 || lane == 16) {
    const int mbase = (lane == 0) ? 0 : 8;
    #pragma unroll
    for (int j = 0; j < 8; ++j) {
      const int t = t0 + mbase + j;
      if (t < T_DIM) {
        const float logit = acc[j] + lb;
        alphas[b * T_DIM + t] = 1.0f / (1.0f + __expf(-logit));
      }
    }
  }
}

// ---------------------------------------------------------------------------
// Kernel 2: scalar CIF scan per batch (integrate recurrence depends only on
// alphas). Produces cur/rem/slot scratch + token_num + cif_peak.
// ---------------------------------------------------------------------------
__global__ __launch_bounds__(32) void cif_scan_kernel(
    const float* __restrict__ alphas, float* __restrict__ cur_a,
    float* __restrict__ rem_a, int* __restrict__ slot_a,
    float* __restrict__ token_num, float* __restrict__ cif_peak)
{
  const int b = threadIdx.x;
  if (b >= B_DIM) return;
  float integ = 0.0f, peak = 0.0f, tok = 0.0f;
  int pos = 0;
  for (int t = 0; t < T_DIM; ++t) {
    const int idx = b * T_DIM + t;
    const float alpha = alphas[idx];
    tok += alpha;
    const float ip = integ + alpha;          // recorded "fires" value
    const bool fire = (ip >= 1.0f);
    const float cur = fire ? (1.0f - integ) : alpha;
    cur_a[idx]  = cur;
    rem_a[idx]  = alpha - cur;
    slot_a[idx] = fire ? pos : -1;
    if (fire) pos++;
    peak += ip;
    cif_peak[idx] = peak;
    integ = fire ? (ip - 1.0f) : ip;
  }
  token_num[b] = tok;
}

// ---------------------------------------------------------------------------
// Kernel 3: zero-fill acoustic_embeds (only fired rows get overwritten).
// ---------------------------------------------------------------------------
__global__ void zero_kernel(float* __restrict__ out, int n4)
{
  int i = blockIdx.x * blockDim.x + threadIdx.x;
  const int stride = gridDim.x * blockDim.x;
  v4f z = {};
  for (; i < n4; i += stride) *(v4f*)(out + 4 * (size_t)i) = z;
}

// ---------------------------------------------------------------------------
// Kernel 4: frame accumulation + scatter of fired frames, with double-buffered
// GLOBAL_LOAD_ASYNC_TO_LDS_B128 streaming of hidden rows (tiles are contiguous
// 24KB chunks). s_wait_asynccnt 3 leaves next tile's loads in flight.
// ---------------------------------------------------------------------------
__device__ __forceinline__ void async_issue_tile(const float* gsrc,
                                                 unsigned lds_base, int tid)
{
  #pragma unroll
  for (int i = 0; i < 3; ++i) {
    unsigned lds = lds_base + (unsigned)(i * 8192 + tid * 16);
    unsigned long long gaddr =
        (unsigned long long)(const void*)(gsrc + i * 2048 + tid * 4);
    asm volatile("global_load_async_to_lds_b128 %0, %1, off"
                 :: "v"(lds), "v"(gaddr) : "memory");
  }
}

__global__ __launch_bounds__(512) void cif_frames_kernel(
    const float* __restrict__ hidden, const float* __restrict__ cur_a,
    const float* __restrict__ rem_a, const int* __restrict__ slot_a,
    float* __restrict__ out)
{
  __shared__ float buf[2 * TT * C_DIM];     // 48 KB double buffer

  const int b   = blockIdx.x;
  const int tid = threadIdx.x;              // 0..511, = channel c
  const float* hb = hidden + (size_t)b * T_DIM * C_DIM;
  float* ob = out + (size_t)b * T_DIM * C_DIM;
  const unsigned lds0 = (unsigned)(unsigned long long)(const void*)&buf[0];

  // prologue: tiles 0 and 1 in flight
  async_issue_tile(hb, lds0, tid);
  async_issue_tile(hb + TT * C_DIM, lds0 + TT * C_DIM * 4, tid);

  float frame = 0.0f;
  for (int k = 0; k < NTILES; ++k) {
    asm volatile("s_wait_asynccnt 0x3" ::: "memory");   // tile k landed (mine)
    __syncthreads();                                     // tile k landed (all waves)

    const float* hbuf = &buf[(k & 1) * TT * C_DIM];
    #pragma unroll
    for (int r = 0; r < TT; ++r) {
      const int t   = k * TT + r;
      const int idx = b * T_DIM + t;
      const float h  = hbuf[r * C_DIM + tid];
      const float cu = cur_a[idx];
      const float fo = frame + cu * h;
      const int   sl = slot_a[idx];
      if (sl >= 0) {
        ob[(size_t)sl * C_DIM + tid] = fo;
        frame = rem_a[idx] * h;
      } else {
        frame = fo;
      }
    }
    __syncthreads();                                     // done reading buf[k&1]
    if (k + 2 < NTILES) {
      async_issue_tile(hb + (size_t)(k + 2) * TT * C_DIM,
                       lds0 + (unsigned)((k & 1) * TT * C_DIM * 4), tid);
    }
  }
}

// ---------------------------------------------------------------------------
extern "C" void kernel_launch(void* const* d_in, const int* in_sizes, int n_in,
                              void* d_out, int out_size, void* d_ws, size_t ws_size,
                              hipStream_t stream)
{
  const float* hidden = (const float*)d_in[0];
  const float* conv_w = (const float*)d_in[1];
  const float* conv_b = (const float*)d_in[2];
  const float* lin_w  = (const float*)d_in[3];
  const float* lin_b  = (const float*)d_in[4];

  float* out_f = (float*)d_out;
  const size_t ACC_N  = (size_t)B_DIM * T_DIM * C_DIM;   // 12,288,000
  float* acoustic  = out_f;
  float* token_num = out_f + ACC_N;
  float* alphas    = out_f + ACC_N + B_DIM;
  float* cif_peak  = out_f + ACC_N + B_DIM + (size_t)B_DIM * T_DIM;

  const size_t BT = (size_t)B_DIM * T_DIM;               // 24,000
  float* cur_a  = (float*)d_ws;
  float* rem_a  = cur_a + BT;
  int*   slot_a = (int*)(rem_a + BT);

  // 1) alphas via WMMA-fused conv+GEMV+sigmoid
  alphas_kernel<<<B_DIM * TILES_PER_B, 32, 0, stream>>>(
      hidden, conv_w, conv_b, lin_w, lin_b, alphas);

  // 2) scalar CIF scan
  cif_scan_kernel<<<1, 32, 0, stream>>>(alphas, cur_a, rem_a, slot_a,
                                        token_num, cif_peak);

  // 3) zero acoustic region
  zero_kernel<<<4096, 256, 0, stream>>>(acoustic, (int)(ACC_N / 4));

  // 4) frame accumulation + scatter with async LDS streaming
  cif_frames_kernel<<<B_DIM, 512, 0, stream>>>(hidden, cur_a, rem_a, slot_a,
                                               acoustic);
}